// AdaptiveCompressionLayer_63883343560888
// MI455X (gfx1250) — compile-verified
//
#include <hip/hip_runtime.h>

#define H       768
#define S_TOTAL 65536
#define MT      32      // rows per workgroup
#define XS      772     // padded LDS row stride for X / y (floats): 772%64==4 -> conflict-free
#define HS      708     // padded LDS row stride for h (floats), >= max padded hc (704)
#define LN_EPS  1e-5f

typedef __attribute__((ext_vector_type(2))) float v2f;
typedef __attribute__((ext_vector_type(8))) float v8f;

__device__ __forceinline__ int imin(int a, int b) { return a < b ? a : b; }

// D = A(16x4) * B(4x16) + C, full fp32 (exact vs. reference)
__device__ __forceinline__ v8f wmma4(v2f a, v2f b, v8f c) {
    return __builtin_amdgcn_wmma_f32_16x16x4_f32(
        /*neg_a=*/false, a, /*neg_b=*/false, b,
        /*c_mod=*/(short)0, c, /*reuse_a=*/false, /*reuse_b=*/false);
}

__global__ __launch_bounds__(256, 1)
void acl_fused_kernel(const float* __restrict__ X, const float* __restrict__ scores,
                      const float* __restrict__ Wc0, const float* __restrict__ bc0,
                      const float* __restrict__ Wd0, const float* __restrict__ bd0,
                      const float* __restrict__ Wc1, const float* __restrict__ bc1,
                      const float* __restrict__ Wd1, const float* __restrict__ bd1,
                      const float* __restrict__ Wc2, const float* __restrict__ bc2,
                      const float* __restrict__ Wd2, const float* __restrict__ bd2,
                      const float* __restrict__ gamma, const float* __restrict__ beta,
                      float* __restrict__ out)
{
    extern __shared__ float smem[];
    float* lx   = smem;                       // MT x XS  (X tile, later y tile)
    float* lh   = smem + MT * XS;             // MT x HS  (masked bottleneck tile)
    int*   lsel = (int*)(smem + MT * XS + MT * HS);  // MT routing ids

    const int tid  = threadIdx.x;
    const int lane = tid & 31;
    const int wave = tid >> 5;
    const int rowBase = blockIdx.x * MT;

    const int nlo   = lane & 15;        // N (or M for A-frags)
    const int lhi   = lane >> 4;        // half-wave id
    const int khalf = lhi * 2;          // K base within a 4-wide K slab

    // ---------------- stage X tile (32 x 768) into LDS ----------------
    for (int i = tid; i < MT * (H / 4); i += 256) {
        const int r  = i / (H / 4);
        const int c4 = i % (H / 4);
        const float4 v = ((const float4*)(X + (size_t)(rowBase + r) * H))[c4];
        *(float4*)(lx + r * XS + c4 * 4) = v;
    }
    if (tid < MT) {
        const float s = scores[rowBase + tid];
        lsel[tid] = (s > 0.8f) ? 0 : (s > 0.4f) ? 1 : 2;
    }
    __syncthreads();

    // ---------------- persistent y accumulators ----------------
    // y tile is 32x768 = 2 x 48 (16x16) tiles; wave w owns mt = w>>2, nt in [ (w&3)*12, +12 )
    v8f acc[12];
#pragma unroll
    for (int i = 0; i < 12; ++i)
#pragma unroll
        for (int v = 0; v < 8; ++v) acc[i][v] = 0.0f;

    const int ymt  = wave >> 2;
    const int ynt0 = (wave & 3) * 12;
    const int hmt  = wave >> 2;         // fixed m-tile per wave for the h-phase
    const int hnt0 = wave & 3;          // n-tiles strided by 4 across the 4 waves sharing hmt

    const float* WcA[3] = {Wc0, Wc1, Wc2};
    const float* BcA[3] = {bc0, bc1, bc2};
    const float* WdA[3] = {Wd0, Wd1, Wd2};
    const int    hcA[3]  = {691, 537, 76};
    const int    hcpA[3] = {704, 544, 80};

#pragma unroll
    for (int e = 0; e < 3; ++e) {
        const float* wc  = WcA[e];
        const float* bc  = BcA[e];
        const float* wd  = WdA[e];
        const int    hc  = hcA[e];
        const int    hcp = hcpA[e];
        const int    nNT = hcp >> 4;    // n-tiles in h

        __syncthreads();   // h buffer free (previous expert's y-phase done)

        // ---- h = routing_mask .* (X @ Wc + bc), stored to LDS ----
        // Each wave: fixed m-tile, n-tiles processed in pairs sharing one A-frag load.
        const int arowX = (hmt * 16 + nlo) * XS;
        for (int nt = hnt0; nt < nNT; nt += 8) {
            const bool pair = (nt + 4) < nNT;        // wave-uniform
            const int colA = nt * 16 + nlo;
            const int colB = (nt + 4) * 16 + nlo;
            const int cA = imin(colA, hc - 1);       // clamp: OOR cols masked to 0 at store
            const int cB = imin(colB, hc - 1);
            v8f h0, h1;
            const float bA = bc[cA];
            const float bB = bc[cB];
#pragma unroll
            for (int v = 0; v < 8; ++v) { h0[v] = bA; h1[v] = bB; }

            if (pair) {
                for (int k = 0; k < H; k += 4) {
                    const float* ap = lx + arowX + k + khalf;
                    v2f a; a.x = ap[0]; a.y = ap[1];            // one ds_load_b64
                    const float* wp = wc + (k + khalf) * hc;
                    v2f b0; b0.x = wp[cA]; b0.y = wp[hc + cA];
                    v2f b1; b1.x = wp[cB]; b1.y = wp[hc + cB];
                    h0 = wmma4(a, b0, h0);
                    h1 = wmma4(a, b1, h1);
                }
            } else {
                for (int k = 0; k < H; k += 4) {
                    const float* ap = lx + arowX + k + khalf;
                    v2f a; a.x = ap[0]; a.y = ap[1];
                    const float* wp = wc + (k + khalf) * hc;
                    v2f b0; b0.x = wp[cA]; b0.y = wp[hc + cA];
                    h0 = wmma4(a, b0, h0);
                }
            }
#pragma unroll
            for (int v = 0; v < 8; ++v) {
                const int m = hmt * 16 + v + 8 * lhi;
                const bool rowok = (lsel[m] == e);
                lh[m * HS + colA] = (rowok && colA < hc) ? h0[v] : 0.0f;
                if (pair) lh[m * HS + colB] = (rowok && colB < hc) ? h1[v] : 0.0f;
            }
        }
        __syncthreads();   // h ready

        // ---- y += h @ Wd : K-outer, one A-frag feeds all 12 output tiles ----
        const int arowH = (ymt * 16 + nlo) * HS;
        for (int k = 0; k < hcp; k += 4) {
            const float* ap = lh + arowH + k + khalf;
            v2f a; a.x = ap[0]; a.y = ap[1];                    // one ds_load_b64 / 12 wmmas
            const int krow = k + khalf;
            // h rows >= hc are zero, so clamped Wd rows contribute exactly 0
            const float* w0 = wd + imin(krow,     hc - 1) * H;
            const float* w1 = wd + imin(krow + 1, hc - 1) * H;
#pragma unroll
            for (int i = 0; i < 12; ++i) {
                const int col = (ynt0 + i) * 16 + nlo;
                v2f b; b.x = w0[col]; b.y = w1[col];
                acc[i] = wmma4(a, b, acc[i]);
            }
        }
    }

    // ---- add routed decoder bias, spill y into lx (X is dead now) ----
#pragma unroll
    for (int i = 0; i < 12; ++i) {
        const int col = (ynt0 + i) * 16 + nlo;
#pragma unroll
        for (int v = 0; v < 8; ++v) {
            const int m = ymt * 16 + v + 8 * lhi;
            const int sel = lsel[m];
            const float* bdp = (sel == 0) ? bd0 : (sel == 1) ? bd1 : bd2;
            lx[m * XS + col] = acc[i][v] + bdp[col];
        }
    }
    __syncthreads();

    // ---- LayerNorm per row: one wave32 per row (each wave does 4 rows) ----
    for (int r = wave; r < MT; r += 8) {
        const float* yr = lx + r * XS;
        float s = 0.0f;
        for (int c = lane; c < H; c += 32) s += yr[c];
#pragma unroll
        for (int off = 16; off > 0; off >>= 1) s += __shfl_xor(s, off, 32);
        const float mu = s * (1.0f / H);

        float vs = 0.0f;
        for (int c = lane; c < H; c += 32) { const float d = yr[c] - mu; vs += d * d; }
#pragma unroll
        for (int off = 16; off > 0; off >>= 1) vs += __shfl_xor(vs, off, 32);
        const float inv = rsqrtf(vs * (1.0f / H) + LN_EPS);

        float* orow = out + (size_t)(rowBase + r) * H;
        for (int c = lane; c < H; c += 32)
            orow[c] = (yr[c] - mu) * inv * gamma[c] + beta[c];
    }
}

extern "C" void kernel_launch(void* const* d_in, const int* in_sizes, int n_in,
                              void* d_out, int out_size, void* d_ws, size_t ws_size,
                              hipStream_t stream) {
    const float* X      = (const float*)d_in[0];
    const float* scores = (const float*)d_in[1];
    const float* Wc0 = (const float*)d_in[2];
    const float* bc0 = (const float*)d_in[3];
    const float* Wd0 = (const float*)d_in[4];
    const float* bd0 = (const float*)d_in[5];
    const float* Wc1 = (const float*)d_in[6];
    const float* bc1 = (const float*)d_in[7];
    const float* Wd1 = (const float*)d_in[8];
    const float* bd1 = (const float*)d_in[9];
    const float* Wc2 = (const float*)d_in[10];
    const float* bc2 = (const float*)d_in[11];
    const float* Wd2 = (const float*)d_in[12];
    const float* bd2 = (const float*)d_in[13];
    const float* gamma = (const float*)d_in[14];
    const float* beta  = (const float*)d_in[15];
    float* out = (float*)d_out;

    const size_t shmem = (size_t)(MT * XS + MT * HS + MT) * sizeof(float); // ~185 KB < 320 KB/WGP
    (void)hipFuncSetAttribute((const void*)acl_fused_kernel,
                              hipFuncAttributeMaxDynamicSharedMemorySize, (int)shmem);

    dim3 grid(S_TOTAL / MT);   // 2048 workgroups
    dim3 block(256);           // 8 wave32
    acl_fused_kernel<<<grid, block, shmem, stream>>>(
        X, scores, Wc0, bc0, Wd0, bd0, Wc1, bc1, Wd1, bd1,
        Wc2, bc2, Wd2, bd2, gamma, beta, out);
}